// JointRelationModule_19713899889182
// MI455X (gfx1250) — compile-verified
//
#include <hip/hip_runtime.h>
#include <hip/hip_bf16.h>

typedef float v2f __attribute__((ext_vector_type(2)));
typedef float v8f __attribute__((ext_vector_type(8)));

#define P          512
#define KC         17
#define HW         4096
#define NUM_IMAGES 64
#define TILE       256
#define ROWS       260   /* TILE + 4 pad -> conflict-free WMMA b64 LDS reads */
#define G_STRIDE   320   /* [0..288]=17x17, [289..305]=vec17 */
#define SC_STRIDE  304

// ---------------------------------------------------------------------------
// Pass 1: per-person Gram matrix G = X X^T (17x17) and row sums s = X*1.
// 16x16 block of G via V_WMMA_F32_16X16X4_F32 (A and B are the SAME regs),
// row/col 16 and s via VALU.
// ---------------------------------------------------------------------------
__global__ __launch_bounds__(256) void k_gram(const float* __restrict__ X0,
                                              float* __restrict__ Gws) {
  __shared__ float xt[KC * ROWS];
  __shared__ float wred[8 * 256];
  __shared__ float g16s[KC];
  __shared__ float ssh[KC];
  const int p    = blockIdx.x;
  const int tid  = threadIdx.x;
  const int lane = tid & 31;
  const int wv   = tid >> 5;
  const float* X = X0 + (size_t)p * KC * HW;

  if (tid < KC) { g16s[tid] = 0.f; ssh[tid] = 0.f; }

  v8f acc = {};
  float g16a[KC], sa[KC];
#pragma unroll
  for (int j = 0; j < KC; ++j) { g16a[j] = 0.f; sa[j] = 0.f; }

  const int ch   = lane & 15;        // A: M-row == B: N-col == channel
  const int koff = (lane >> 4) * 2;  // K pair held by this half-wave

  for (int t = 0; t < HW / TILE; ++t) {
    __syncthreads();
    // Coalesced float4 staging of a 17 x 256-pixel slab into padded LDS.
    for (int f = tid * 4; f < KC * TILE; f += 256 * 4) {
      const int row = f / TILE, col = f % TILE;
      const float4 v = *(const float4*)(X + (size_t)row * HW + t * TILE + col);
      *(float4*)(&xt[row * ROWS + col]) = v;
    }
    __syncthreads();
    // Each wave accumulates 8 K-chunks of 4 pixels: G16x16 += A * A.
#pragma unroll
    for (int q = 0; q < 8; ++q) {
      const int kc = (wv * 8 + q) * 4;
      v2f a = *(const v2f*)(&xt[ch * ROWS + kc + koff]);
      acc = __builtin_amdgcn_wmma_f32_16x16x4_f32(false, a, false, a,
                                                  (short)0, acc, false, false);
    }
    // Channel-16 row of G and the row sums s (thread <-> pixel).
    const float x16 = xt[16 * ROWS + tid];
#pragma unroll
    for (int j = 0; j < KC; ++j) {
      const float xj = xt[j * ROWS + tid];
      g16a[j] += x16 * xj;
      sa[j]   += xj;
    }
  }

  // Wave-reduce scalar accumulators, then LDS atomics from lane 0.
#pragma unroll
  for (int j = 0; j < KC; ++j) {
    float g = g16a[j], s = sa[j];
    for (int o = 16; o > 0; o >>= 1) {
      g += __shfl_down(g, o, 32);
      s += __shfl_down(s, o, 32);
    }
    if (lane == 0) { atomicAdd(&g16s[j], g); atomicAdd(&ssh[j], s); }
  }

  // Cross-wave reduction of the 8 WMMA accumulators.
#pragma unroll
  for (int r = 0; r < 8; ++r) wred[wv * 256 + lane * 8 + r] = acc[r];
  __syncthreads();

  float* Gp = Gws + (size_t)p * G_STRIDE;
  {
    const int l = tid >> 3, r = tid & 7;
    float sum = 0.f;
#pragma unroll
    for (int w2 = 0; w2 < 8; ++w2) sum += wred[w2 * 256 + l * 8 + r];
    const int m = r + 8 * (l >> 4), n = l & 15;   // documented C/D layout
    Gp[m * KC + n] = sum;
  }
  if (tid < KC) {
    const float g = g16s[tid];
    Gp[16 * KC + tid] = g;
    Gp[tid * KC + 16] = g;
    Gp[289 + tid]     = ssh[tid];
  }
}

// ---------------------------------------------------------------------------
// Pass 2a: scores[p] = (Wq G Wk^T + (Wq s) bk^T + bq (Wk s)^T + hw*bq bk^T)/64
// ---------------------------------------------------------------------------
__global__ __launch_bounds__(320) void k_scores(const float* __restrict__ Gws,
                                                const float* __restrict__ Wq,
                                                const float* __restrict__ bq,
                                                const float* __restrict__ Wk,
                                                const float* __restrict__ bk,
                                                float* __restrict__ Sws) {
  __shared__ float Gl[289], T1[289], Wql[289], Wkl[289], sl[17], bql[17], bkl[17];
  const int p = blockIdx.x, t = threadIdx.x;
  const float* Gp = Gws + (size_t)p * G_STRIDE;
  for (int i = t; i < 289; i += 320) { Gl[i] = Gp[i]; Wql[i] = Wq[i]; Wkl[i] = Wk[i]; }
  if (t < 17) { sl[t] = Gp[289 + t]; bql[t] = bq[t]; bkl[t] = bk[t]; }
  __syncthreads();
  if (t < 289) {
    const int i = t / 17, b = t % 17;
    float a = 0.f;
    for (int k2 = 0; k2 < 17; ++k2) a += Wql[i * 17 + k2] * Gl[k2 * 17 + b];
    T1[t] = a;
  }
  __syncthreads();
  if (t < 289) {
    const int i = t / 17, j = t % 17;
    float sc = 0.f, u = 0.f, tv = 0.f;
    for (int b = 0; b < 17; ++b) sc += T1[i * 17 + b] * Wkl[j * 17 + b];
    for (int a = 0; a < 17; ++a) { u += Wql[i * 17 + a] * sl[a]; tv += Wkl[j * 17 + a] * sl[a]; }
    sc += u * bkl[j] + bql[i] * tv + (float)HW * bql[i] * bkl[j];
    Sws[(size_t)p * SC_STRIDE + t] = sc * (1.0f / 64.0f);
  }
}

// ---------------------------------------------------------------------------
// Pass 2b: per-image segment softmax over persons, then A' = attn*Wv + I and
// c = attn*bv per person.  One block per image; imgid is sorted.
// ---------------------------------------------------------------------------
__global__ __launch_bounds__(320) void k_softmax(const float* __restrict__ Sws,
                                                 const int* __restrict__ imgid,
                                                 const float* __restrict__ Wv,
                                                 const float* __restrict__ bv,
                                                 float* __restrict__ Aws) {
  __shared__ float Wvl[289], bvl[17], attn[289];
  __shared__ int sp0, sp1;
  const int g = blockIdx.x, t = threadIdx.x;
  for (int i = t; i < 289; i += 320) Wvl[i] = Wv[i];
  if (t < 17) bvl[t] = bv[t];
  if (t == 0) {
    int p0 = 0, p1 = 0;
    for (int p = 0; p < P; ++p) { const int id = imgid[p]; p0 += (id < g); p1 += (id <= g); }
    sp0 = p0; sp1 = p1;
  }
  __syncthreads();
  const int p0 = sp0, p1 = sp1;
  const int i = t / 17, j = t % 17;

  float m = -3.0e38f, ssum = 0.f;
  if (t < 289) {
    for (int p = p0; p < p1; ++p) m = fmaxf(m, Sws[(size_t)p * SC_STRIDE + t]);
    for (int p = p0; p < p1; ++p) ssum += __expf(Sws[(size_t)p * SC_STRIDE + t] - m);
  }
  for (int p = p0; p < p1; ++p) {
    __syncthreads();
    if (t < 289) attn[t] = __expf(Sws[(size_t)p * SC_STRIDE + t] - m) / ssum;
    __syncthreads();
    float* Ap = Aws + (size_t)p * G_STRIDE;
    if (t < 289) {
      float a = 0.f;
      for (int jj = 0; jj < 17; ++jj) a += attn[i * 17 + jj] * Wvl[jj * 17 + j];
      if (i == j) a += 1.0f;       // fold the residual into A'
      Ap[t] = a;
    }
    if (t < 17) {
      float c = 0.f;
      for (int jj = 0; jj < 17; ++jj) c += attn[t * 17 + jj] * bvl[jj];
      Ap[289 + t] = c;
    }
  }
}

// ---------------------------------------------------------------------------
// Pass 3: out = relu(A' * X + c).  A' held in WMMA A-registers (5 K-chunks,
// chunk 4 zero-padded past col 16); row-16 output done with VALU + cross-half
// shuffles.  One block per person, 8 waves x 32 pixel-tiles.
// ---------------------------------------------------------------------------
__global__ __launch_bounds__(256) void k_out(const float* __restrict__ X0,
                                             const float* __restrict__ Aws,
                                             float* __restrict__ Out) {
  __shared__ float Al[306];
  const int p = blockIdx.x, tid = threadIdx.x;
  const int lane = tid & 31, wv = tid >> 5;
  const float* Ap = Aws + (size_t)p * G_STRIDE;
  for (int i = tid; i < 306; i += 256) Al[i] = Ap[i];
  __syncthreads();

  const int n = lane & 15;
  const int h = lane >> 4;
  const int m = n;                 // A-matrix row per lane
  v2f a[5];
#pragma unroll
  for (int c = 0; c < 5; ++c) {
    const int c0 = c * 4 + 2 * h, c1 = c0 + 1;
    a[c].x = (c0 < 17) ? Al[m * 17 + c0] : 0.f;
    a[c].y = (c1 < 17) ? Al[m * 17 + c1] : 0.f;
  }
  float a16[17];
#pragma unroll
  for (int k2 = 0; k2 < 17; ++k2) a16[k2] = Al[16 * 17 + k2];
  float cc[8];
#pragma unroll
  for (int r = 0; r < 8; ++r) cc[r] = Al[289 + r + 8 * h];
  const float c16 = Al[289 + 16];

  const float* X = X0 + (size_t)p * KC * HW;
  float*       O = Out + (size_t)p * KC * HW;

  for (int tI = wv; tI < HW / 16; tI += 8) {
    const int pix = tI * 16 + n;
    v2f b0, b1, b2, b3, b4;
    b0.x = X[(size_t)(0  + 2 * h) * HW + pix]; b0.y = X[(size_t)(1  + 2 * h) * HW + pix];
    b1.x = X[(size_t)(4  + 2 * h) * HW + pix]; b1.y = X[(size_t)(5  + 2 * h) * HW + pix];
    b2.x = X[(size_t)(8  + 2 * h) * HW + pix]; b2.y = X[(size_t)(9  + 2 * h) * HW + pix];
    b3.x = X[(size_t)(12 + 2 * h) * HW + pix]; b3.y = X[(size_t)(13 + 2 * h) * HW + pix];
    const float x16 = X[(size_t)16 * HW + pix];
    b4.x = h ? 0.f : x16; b4.y = 0.f;

    v8f acc = {};
    acc = __builtin_amdgcn_wmma_f32_16x16x4_f32(false, a[0], false, b0, (short)0, acc, false, false);
    acc = __builtin_amdgcn_wmma_f32_16x16x4_f32(false, a[1], false, b1, (short)0, acc, false, false);
    acc = __builtin_amdgcn_wmma_f32_16x16x4_f32(false, a[2], false, b2, (short)0, acc, false, false);
    acc = __builtin_amdgcn_wmma_f32_16x16x4_f32(false, a[3], false, b3, (short)0, acc, false, false);
    acc = __builtin_amdgcn_wmma_f32_16x16x4_f32(false, a[4], false, b4, (short)0, acc, false, false);

    // Output channel 16: gather the other half-wave's rows via shuffles.
    float o16 = c16 + a16[16] * x16;
    o16 += a16[0]  * b0.x + a16[1]  * b0.y + a16[2]  * __shfl(b0.x, lane + 16, 32) + a16[3]  * __shfl(b0.y, lane + 16, 32);
    o16 += a16[4]  * b1.x + a16[5]  * b1.y + a16[6]  * __shfl(b1.x, lane + 16, 32) + a16[7]  * __shfl(b1.y, lane + 16, 32);
    o16 += a16[8]  * b2.x + a16[9]  * b2.y + a16[10] * __shfl(b2.x, lane + 16, 32) + a16[11] * __shfl(b2.y, lane + 16, 32);
    o16 += a16[12] * b3.x + a16[13] * b3.y + a16[14] * __shfl(b3.x, lane + 16, 32) + a16[15] * __shfl(b3.y, lane + 16, 32);

#pragma unroll
    for (int r = 0; r < 8; ++r) {
      const float v = fmaxf(acc[r] + cc[r], 0.f);
      O[(size_t)(r + 8 * h) * HW + pix] = v;
    }
    if (h == 0) O[(size_t)16 * HW + pix] = fmaxf(o16, 0.f);
  }
}

extern "C" void kernel_launch(void* const* d_in, const int* in_sizes, int n_in,
                              void* d_out, int out_size, void* d_ws, size_t ws_size,
                              hipStream_t stream) {
  (void)in_sizes; (void)n_in; (void)out_size; (void)ws_size;
  const float* X   = (const float*)d_in[0];
  const float* Wk  = (const float*)d_in[1];
  const float* bk  = (const float*)d_in[2];
  const float* Wq  = (const float*)d_in[3];
  const float* bq  = (const float*)d_in[4];
  const float* Wv  = (const float*)d_in[5];
  const float* bv  = (const float*)d_in[6];
  const int* imgid = (const int*)d_in[7];
  float* out = (float*)d_out;

  float* Gws = (float*)d_ws;                       // 512*320 floats
  float* Sws = Gws + (size_t)P * G_STRIDE;         // 512*304 floats
  float* Aws = Sws + (size_t)P * SC_STRIDE;        // 512*320 floats (~1.9MB total)

  k_gram   <<<P,          256, 0, stream>>>(X, Gws);
  k_scores <<<P,          320, 0, stream>>>(Gws, Wq, bq, Wk, bk, Sws);
  k_softmax<<<NUM_IMAGES, 320, 0, stream>>>(Sws, imgid, Wv, bv, Aws);
  k_out    <<<P,          256, 0, stream>>>(X, Aws, out);
}